// SimpleGNN_28080496181754
// MI455X (gfx1250) — compile-verified
//
#include <hip/hip_runtime.h>
#include <hip/hip_bf16.h>
#include <math.h>

// ---------------------------------------------------------------------------
// GCN forward for gfx1250 (MI455X): f32 WMMA GEMMs + L2-resident atomic scatter
// ---------------------------------------------------------------------------

typedef float v2f __attribute__((ext_vector_type(2)));
typedef float v8f __attribute__((ext_vector_type(8)));

#define DH 128  // hidden/input width (reference: d_in = d_h = 128)
#define DO 10   // output classes

// ---- generic fill -----------------------------------------------------------
__global__ void k_fill(float* __restrict__ p, float v, long long n) {
  long long i = (long long)blockIdx.x * blockDim.x + threadIdx.x;
  if (i < n) p[i] = v;
}

// ---- degree accumulation (deg pre-initialized to 1.0 for self loops) --------
__global__ void k_deg(const int* __restrict__ dst, float* __restrict__ deg, int E) {
  int e = blockIdx.x * blockDim.x + threadIdx.x;
  if (e < E) atomicAdd(&deg[dst[e]], 1.0f);
}

// ---- deg -> 1/sqrt(deg) in place (deg >= 1 always due to self loops) --------
__global__ void k_dis(float* __restrict__ deg, int n) {
  int i = blockIdx.x * blockDim.x + threadIdx.x;
  if (i < n) deg[i] = rsqrtf(deg[i]);
}

// ---- GEMM: T[n x 128] = A[n x 128] @ W[128 x 128] via V_WMMA_F32_16X16X4_F32
// Block = 256 threads = 8 wave32s. Block owns a 16-row M tile; wave w owns
// N columns [16w, 16w+16). A tile staged in LDS (stride 132 -> conflict free).
__global__ __launch_bounds__(256) void k_gemm(const float* __restrict__ A,
                                              const float* __restrict__ W,
                                              float* __restrict__ T, int n) {
  __shared__ float As[16][132];
  const int m0  = blockIdx.x * 16;
  const int tid = threadIdx.x;

  // Cooperative A-tile load: 16 rows x 128 cols = 512 float4s, coalesced.
  for (int i = tid; i < 16 * 32; i += 256) {
    int r  = i >> 5;          // row 0..15
    int c4 = (i & 31) * 4;    // col group
    float4 v = make_float4(0.f, 0.f, 0.f, 0.f);
    if (m0 + r < n) v = *(const float4*)(A + (size_t)(m0 + r) * DH + c4);
    As[r][c4 + 0] = v.x; As[r][c4 + 1] = v.y;
    As[r][c4 + 2] = v.z; As[r][c4 + 3] = v.w;
  }
  __syncthreads();

  const int wave = tid >> 5;       // 0..7 -> N tile
  const int lane = tid & 31;
  const int n0   = wave * 16;
  const int half = lane >> 4;      // 0: K+0/1, 1: K+2/3 (ISA A/B lane layout)
  const int l16  = lane & 15;

  v8f c = {};
  for (int kk = 0; kk < DH; kk += 4) {
    const int ka = kk + half * 2;
    // A 16x4: lane l16 -> M=l16; VGPR0=K(ka), VGPR1=K(ka+1)
    v2f a = *(const v2f*)&As[l16][ka];
    // B 4x16: lane l16 -> N=n0+l16; rows ka, ka+1 of W (row-major [K][N])
    v2f b;
    b.x = W[(size_t)(ka + 0) * DH + n0 + l16];
    b.y = W[(size_t)(ka + 1) * DH + n0 + l16];
    c = __builtin_amdgcn_wmma_f32_16x16x4_f32(
        /*neg_a=*/false, a, /*neg_b=*/false, b,
        /*c_mod=*/(short)0, c, /*reuse_a=*/false, /*reuse_b=*/false);
  }

  // C/D layout: VGPR r -> M = r (lanes 0-15) / r+8 (lanes 16-31); N = l16.
#pragma unroll
  for (int r = 0; r < 8; ++r) {
    int m = m0 + r + half * 8;
    if (m < n) T[(size_t)m * DH + n0 + l16] = c[r];
  }
}

// ---- edge scatter: agg[dst] += T[src] * dis[src]*dis[dst] -------------------
// One wave per edge per iteration; lane handles 4 channels (float4 gather +
// 4x global_atomic_add_f32). Prefetch next edge's source row into cache.
__global__ __launch_bounds__(256) void k_scatter(const float* __restrict__ T,
                                                 const int* __restrict__ src,
                                                 const int* __restrict__ dst,
                                                 const float* __restrict__ dis,
                                                 float* __restrict__ agg, int E) {
  const int lane = threadIdx.x & 31;
  const int wid  = (blockIdx.x * blockDim.x + threadIdx.x) >> 5;
  const int nw   = (gridDim.x * blockDim.x) >> 5;
  for (int e = wid; e < E; e += nw) {
    int en = e + nw;
    if (en < E) {
      int sn = src[en];
      __builtin_prefetch(T + (size_t)sn * DH + lane * 4, 0, 3);  // global_prefetch_b8
    }
    int s = src[e], d = dst[e];
    float nrm = dis[s] * dis[d];
    float4 v = *(const float4*)(T + (size_t)s * DH + lane * 4);
    float* out = agg + (size_t)d * DH + lane * 4;
    atomicAdd(out + 0, v.x * nrm);
    atomicAdd(out + 1, v.y * nrm);
    atomicAdd(out + 2, v.z * nrm);
    atomicAdd(out + 3, v.w * nrm);
  }
}

// ---- self-loop + bias + relu, in place on agg -------------------------------
__global__ void k_bias_relu(float* __restrict__ h, const float* __restrict__ T,
                            const float* __restrict__ dis,
                            const float* __restrict__ b, int n) {
  long long t = (long long)blockIdx.x * blockDim.x + threadIdx.x;
  if (t >= (long long)n * DH) return;
  int i = (int)(t >> 7), c = (int)(t & (DH - 1));
  float di = dis[i];
  float v = h[t] + T[t] * di * di + b[c];
  h[t] = v > 0.f ? v : 0.f;
}

// ---- global mean pool (sum + count via atomics) -----------------------------
__global__ void k_pool(const float* __restrict__ h, const int* __restrict__ batch,
                       float* __restrict__ pooled, float* __restrict__ counts, int n) {
  long long t = (long long)blockIdx.x * blockDim.x + threadIdx.x;
  if (t >= (long long)n * DH) return;
  int i = (int)(t >> 7), c = (int)(t & (DH - 1));
  int g = batch[i];
  atomicAdd(pooled + (size_t)g * DH + c, h[t]);
  if (c == 0) atomicAdd(counts + g, 1.0f);
}

// ---- head: mean -> logits -> log_softmax (tiny: one thread per graph) -------
__global__ void k_head(const float* __restrict__ pooled, const float* __restrict__ counts,
                       const float* __restrict__ Wf, const float* __restrict__ bf,
                       float* __restrict__ out, int G) {
  int g = blockIdx.x * blockDim.x + threadIdx.x;
  if (g >= G) return;
  float cnt = counts[g];
  float inv = 1.0f / (cnt > 1.0f ? cnt : 1.0f);
  float logits[DO];
#pragma unroll
  for (int j = 0; j < DO; ++j) logits[j] = bf[j];
  for (int k = 0; k < DH; ++k) {
    float m = pooled[(size_t)g * DH + k] * inv;
#pragma unroll
    for (int j = 0; j < DO; ++j) logits[j] += m * Wf[k * DO + j];
  }
  float mx = logits[0];
#pragma unroll
  for (int j = 1; j < DO; ++j) mx = fmaxf(mx, logits[j]);
  float s = 0.f;
#pragma unroll
  for (int j = 0; j < DO; ++j) s += expf(logits[j] - mx);
  float lse = mx + logf(s);
#pragma unroll
  for (int j = 0; j < DO; ++j) out[(size_t)g * DO + j] = logits[j] - lse;
}

// ---------------------------------------------------------------------------
extern "C" void kernel_launch(void* const* d_in, const int* in_sizes, int n_in,
                              void* d_out, int out_size, void* d_ws, size_t ws_size,
                              hipStream_t stream) {
  const float* x     = (const float*)d_in[0];
  const int*   ei    = (const int*)d_in[1];
  const int*   batch = (const int*)d_in[2];
  const float* Ws[3] = {(const float*)d_in[3], (const float*)d_in[5], (const float*)d_in[7]};
  const float* bs[3] = {(const float*)d_in[4], (const float*)d_in[6], (const float*)d_in[8]};
  const float* Wf    = (const float*)d_in[9];
  const float* bf    = (const float*)d_in[10];
  (void)n_in; (void)ws_size;

  const int n = in_sizes[0] / DH;   // 50000
  const int E = in_sizes[1] / 2;    // 600000
  const int G = out_size / DO;      // 512
  const int* src = ei;              // edge_index[0]
  const int* dst = ei + E;          // edge_index[1]

  // Workspace layout (floats)
  float* ws = (float*)d_ws;
  size_t off = 0;
  float* dis    = ws + off; off += (size_t)n;           // deg -> 1/sqrt(deg)
  float* t      = ws + off; off += (size_t)n * DH;      // GEMM output
  float* bufA   = ws + off; off += (size_t)n * DH;      // ping
  float* bufB   = ws + off; off += (size_t)n * DH;      // pong
  float* pooled = ws + off; off += (size_t)G * DH;
  float* counts = ws + off; off += (size_t)G;

  const int TB = 256;
  const long long nfeat = (long long)n * DH;
  const int gF  = (int)((nfeat + TB - 1) / TB);

  // 1) degree / normalization
  k_fill<<<(n + TB - 1) / TB, TB, 0, stream>>>(dis, 1.0f, n);        // self loops
  k_deg<<<(E + TB - 1) / TB, TB, 0, stream>>>(dst, dis, E);
  k_dis<<<(n + TB - 1) / TB, TB, 0, stream>>>(dis, n);

  // 2) three GCN layers
  const float* cur = x;
  for (int l = 0; l < 3; ++l) {
    float* outb = (l & 1) ? bufB : bufA;
    k_fill<<<gF, TB, 0, stream>>>(outb, 0.0f, nfeat);
    k_gemm<<<(n + 15) / 16, 256, 0, stream>>>(cur, Ws[l], t, n);
    k_scatter<<<8192, TB, 0, stream>>>(t, src, dst, dis, outb, E);
    k_bias_relu<<<gF, TB, 0, stream>>>(outb, t, dis, bs[l], n);
    cur = outb;
  }

  // 3) mean pool + classifier head
  k_fill<<<((long long)G * DH + TB - 1) / TB, TB, 0, stream>>>(pooled, 0.0f, (long long)G * DH);
  k_fill<<<(G + TB - 1) / TB, TB, 0, stream>>>(counts, 0.0f, G);
  k_pool<<<gF, TB, 0, stream>>>(cur, batch, pooled, counts, n);
  k_head<<<(G + TB - 1) / TB, TB, 0, stream>>>(pooled, counts, Wf, bf, (float*)d_out, G);
}